// PCCEVE8_AV_45578192945699
// MI455X (gfx1250) — compile-verified
//
#include <hip/hip_runtime.h>
#include <cstdint>

// Weighted cross-entropy over [B, 8] logits x2 tensors -> scalar mean sum.
// Memory-bound streaming reduction: async global->LDS double-buffered pipeline
// (CDNA5 ASYNCcnt path) + non-temporal hints; deterministic two-pass reduce.

#define NUM_C 8
#define POLARITY_MASK 0xD2   // bit c set iff class c in {1,4,6,7}
#define NTHR 256
#define NBLK 2048

typedef __attribute__((ext_vector_type(4))) float f4;

// Issue one 32-byte row (two b128) from global into LDS asynchronously.
// INST_OFFSET is added to BOTH the LDS and the global address (ISA 08 §4.4),
// so the second half reuses the same VGPR operands with offset:16.
__device__ __forceinline__ void async_row(uint32_t ldsAddr, uint32_t voffBytes,
                                          const float* sbase) {
  asm volatile("global_load_async_to_lds_b128 %0, %1, %2 th:TH_LOAD_NT"
               :: "v"(ldsAddr), "v"(voffBytes), "s"(sbase) : "memory");
  asm volatile("global_load_async_to_lds_b128 %0, %1, %2 offset:16 th:TH_LOAD_NT"
               :: "v"(ldsAddr), "v"(voffBytes), "s"(sbase) : "memory");
}

// Per-row weighted CE given the 8 logits staged in LDS and the true label.
__device__ __forceinline__ float row_wce(const float* rowp, int lab) {
  f4 lo = *(const f4*)(rowp);
  f4 hi = *(const f4*)(rowp + 4);
  float xs[NUM_C] = {lo.x, lo.y, lo.z, lo.w, hi.x, hi.y, hi.z, hi.w};

  float m = xs[0];
  int am = 0;
#pragma unroll
  for (int i = 1; i < NUM_C; ++i) {
    bool gt = xs[i] > m;
    m = gt ? xs[i] : m;
    am = gt ? i : am;
  }
  float s = 0.0f;
#pragma unroll
  for (int i = 0; i < NUM_C; ++i) s += __expf(xs[i] - m);

  float xl = xs[0];
#pragma unroll
  for (int i = 1; i < NUM_C; ++i) xl = (lab == i) ? xs[i] : xl;

  float ce = (m + __logf(s)) - xl;
  int mism = ((POLARITY_MASK >> lab) ^ (POLARITY_MASK >> am)) & 1;
  return ce * (mism ? 1.5f : 1.0f);
}

__global__ __launch_bounds__(NTHR) void wce_partials(
    const float* __restrict__ A, const float* __restrict__ V,
    const int* __restrict__ Y, float* __restrict__ partials, int B) {
  // [buf][tensor][row*8]: 2 * 2 * 256 * 8 floats = 32 KB (of 320 KB/WGP)
  __shared__ __align__(16) float smem[2][2][NTHR * NUM_C];
  __shared__ float red[NTHR];

  const int tid = threadIdx.x;
  const int totalThreads = NBLK * NTHR;
  const int iters = (B + totalThreads - 1) / totalThreads;
  const int blockStart = blockIdx.x * iters * NTHR;

  // LDS byte addresses for this lane's staging slots (low 32 bits of the
  // generic pointer == wave-relative LDS offset).
  const uint32_t ldsBase =
      (uint32_t)(uintptr_t)(&smem[0][0][0]) + (uint32_t)tid * 32u;
  const uint32_t BUFSTRIDE = 2u * NTHR * NUM_C * 4u;  // 16384 B
  const uint32_t TENSTRIDE = NTHR * NUM_C * 4u;       //  8192 B

  // Prime the pipeline: stage 0 into buffer 0.
  {
    int row = blockStart + tid;
    int crow = row < B ? row : (B - 1);
    uint32_t voff = (uint32_t)crow * 32u;
    async_row(ldsBase, voff, A);
    async_row(ldsBase + TENSTRIDE, voff, V);
  }

  float acc = 0.0f;
  for (int it = 0; it < iters; ++it) {
    const int p = it & 1;
    if (it + 1 < iters) {
      // Issue next stage into the other buffer, then wait for current stage
      // (async loads retire in order: <=4 outstanding => first 4 are done).
      int nrow = blockStart + (it + 1) * NTHR + tid;
      int cnrow = nrow < B ? nrow : (B - 1);
      uint32_t voff = (uint32_t)cnrow * 32u;
      uint32_t nbase = ldsBase + (uint32_t)(p ^ 1) * BUFSTRIDE;
      async_row(nbase, voff, A);
      async_row(nbase + TENSTRIDE, voff, V);
      asm volatile("s_wait_asynccnt 4" ::: "memory");
    } else {
      asm volatile("s_wait_asynccnt 0" ::: "memory");
    }

    int row = blockStart + it * NTHR + tid;
    if (row < B) {
      int lab = __builtin_nontemporal_load(Y + row) & 7;
      acc += row_wce(&smem[p][0][tid * NUM_C], lab);
      acc += row_wce(&smem[p][1][tid * NUM_C], lab);
    }
  }

  // Deterministic in-block tree reduction.
  red[tid] = acc;
  __syncthreads();
#pragma unroll
  for (int off = NTHR / 2; off > 0; off >>= 1) {
    if (tid < off) red[tid] += red[tid + off];
    __syncthreads();
  }
  if (tid == 0) partials[blockIdx.x] = red[0];
}

__global__ __launch_bounds__(256) void wce_final(
    const float* __restrict__ partials, int n, float* __restrict__ out, int B) {
  __shared__ double red[256];
  double s = 0.0;
  for (int i = threadIdx.x; i < n; i += 256) s += (double)partials[i];
  red[threadIdx.x] = s;
  __syncthreads();
#pragma unroll
  for (int off = 128; off > 0; off >>= 1) {
    if (threadIdx.x < off) red[threadIdx.x] += red[threadIdx.x + off];
    __syncthreads();
  }
  if (threadIdx.x == 0) out[0] = (float)(red[0] / (double)B);
}

extern "C" void kernel_launch(void* const* d_in, const int* in_sizes, int n_in,
                              void* d_out, int out_size, void* d_ws, size_t ws_size,
                              hipStream_t stream) {
  const float* A = (const float*)d_in[0];   // y_pred_a [B,8] f32
  const float* V = (const float*)d_in[1];   // y_pred_v [B,8] f32
  const int*   Y = (const int*)d_in[2];     // y [B] int
  float* out = (float*)d_out;
  const int B = in_sizes[0] / NUM_C;

  float* partials = (float*)d_ws;           // NBLK floats = 8 KB scratch
  wce_partials<<<NBLK, NTHR, 0, stream>>>(A, V, Y, partials, B);
  wce_final<<<1, 256, 0, stream>>>(partials, NBLK, out, B);
}